// OptimizedCategoryContrastiveLoss_87308095193293
// MI455X (gfx1250) — compile-verified
//
#include <hip/hip_runtime.h>
#include <math.h>

// ---------------------------------------------------------------------------
// Fused contrastive loss for MI455X (gfx1250, wave32).
//   sims = (e/||e||) @ (e/||e||)^T / 0.7 tile-wise via V_WMMA_F32_16X16X4_F32.
//   Per-row masked stats fused in registers. Since |sims/T| <= 1/0.7 < 1.5,
//   logsumexp uses a FIXED shift (1.5): no online max, branch-free update
//   (cndmask + v_exp_f32 + add). B tiles staged to LDS with
//   global_load_async_to_lds_b128 (ASYNCcnt). The Bt base pointer is passed
//   into the asm so the LDS object ESCAPES and the "memory" clobber covers it
//   (otherwise LLVM deletes the ds_loads as reads of never-written memory).
// ---------------------------------------------------------------------------

#define BATCH 8192
#define DIM 256
#define NSPLIT 8                 // column-range splits (grid.y)
#define TC 32                    // columns per LDS tile
#define WAVES 4                  // waves per block
#define THREADS (WAVES * 32)
#define ROWS_PER_BLOCK (WAVES * 16)
#define INV_T (1.0f / 0.7f)
#define SHIFT 1.5f               // fixed logsumexp shift; >= max possible sim/T
#define NEG_INF (-__builtin_inff())

typedef __attribute__((ext_vector_type(2))) float v2f;
typedef __attribute__((ext_vector_type(8))) float v8f;

__global__ void init_acc(float* acc) {
  acc[0] = 0.0f;  // loss sum
  acc[1] = 0.0f;  // valid count
}

// One wave per row: L2-normalize 256 floats.
__global__ __launch_bounds__(256) void normalize_rows(const float* __restrict__ emb,
                                                      float* __restrict__ en) {
  const int row  = (blockIdx.x * blockDim.x + threadIdx.x) >> 5;
  const int lane = threadIdx.x & 31;
  if (row >= BATCH) return;
  const float4* src = (const float4*)(emb + (size_t)row * DIM);
  float4 a = src[lane * 2 + 0];
  float4 b = src[lane * 2 + 1];
  float ss = a.x * a.x + a.y * a.y + a.z * a.z + a.w * a.w +
             b.x * b.x + b.y * b.y + b.z * b.z + b.w * b.w;
#pragma unroll
  for (int off = 1; off < 32; off <<= 1) ss += __shfl_xor(ss, off, 32);
  const float inv = 1.0f / fmaxf(sqrtf(ss), 1e-12f);
  float4* dst = (float4*)(en + (size_t)row * DIM);
  a.x *= inv; a.y *= inv; a.z *= inv; a.w *= inv;
  b.x *= inv; b.y *= inv; b.z *= inv; b.w *= inv;
  dst[lane * 2 + 0] = a;
  dst[lane * 2 + 1] = b;
}

// Main fused kernel: each wave owns 16 rows; block iterates over TC-column
// tiles of its column split. Per tile: 16x32 WMMA f32 accumulation, then
// branch-free lane-private (posmax, negsumexp) updates.
__global__ __launch_bounds__(THREADS) void sim_online(const float* __restrict__ en,
                                                      const int* __restrict__ cats,
                                                      float* __restrict__ pm_o,
                                                      float* __restrict__ ns_o) {
  // +4 pad => row stride 260 dwords: conflict-free ds_load_b64, 16B-aligned rows.
  __shared__ float Bt[TC][DIM + 4];

  const int wave = threadIdx.x >> 5;
  const int lane = threadIdx.x & 31;
  const int half = lane >> 4;      // 0: rows m=0..7 / K+0,1   1: rows m=8..15 / K+2,3
  const int ln   = lane & 15;
  const int ibase = blockIdx.x * ROWS_PER_BLOCK + wave * 16;
  const int j0 = blockIdx.y * (BATCH / NSPLIT);
  const int j1 = j0 + (BATCH / NSPLIT);

  // Escape hatch: generic pointer to the LDS tile, passed into every asm so
  // alias analysis knows the async ops read/write it.
  float* bt_base = &Bt[0][0];

  int rcat[8];
#pragma unroll
  for (int r = 0; r < 8; ++r) rcat[r] = cats[ibase + half * 8 + r];

  float pm[8], ns[8];
#pragma unroll
  for (int r = 0; r < 8; ++r) { pm[r] = NEG_INF; ns[r] = 0.0f; }

  // A-fragment base: lane l holds e[ibase + (l&15)][k + 2*(l>>4) .. +1]
  const float* arow = en + (size_t)(ibase + ln) * DIM + half * 2;

  for (int jt = j0; jt < j1; jt += TC) {
    __syncthreads();
    // Cooperative async B-tile copy: 16 x b128 per thread, LDS offsets are
    // relative to the workgroup LDS base (Bt is the only __shared__ object).
    for (int idx = threadIdx.x; idx < TC * (DIM / 4); idx += THREADS) {
      const int rr = idx / (DIM / 4);
      const int cc = idx - rr * (DIM / 4);
      const unsigned lds_off = (unsigned)((rr * (DIM + 4) + cc * 4) * 4);
      const float* gaddr = en + (size_t)(jt + rr) * DIM + cc * 4;
      asm volatile("global_load_async_to_lds_b128 %0, %1, off"
                   :: "v"(lds_off), "v"(gaddr), "v"(bt_base) : "memory");
    }
    asm volatile("s_wait_asynccnt 0x0" :: "v"(bt_base) : "memory");
    __syncthreads();

    // Prefetch next tile's rows while we compute (global_prefetch_b8).
    if (jt + TC < j1) {
      __builtin_prefetch(en + (size_t)(jt + TC + (threadIdx.x & (TC - 1))) * DIM, 0, 1);
    }

    v8f c0 = {};
    v8f c1 = {};
#pragma unroll 8
    for (int k = 0; k < DIM; k += 4) {
      const v2f a  = *(const v2f*)(arow + k);
      const v2f b0 = *(const v2f*)&Bt[ln][k + half * 2];
      const v2f b1 = *(const v2f*)&Bt[ln + 16][k + half * 2];
      c0 = __builtin_amdgcn_wmma_f32_16x16x4_f32(false, a, false, b0, (short)0, c0,
                                                 false, false);
      c1 = __builtin_amdgcn_wmma_f32_16x16x4_f32(false, a, false, b1, (short)0, c1,
                                                 false, false);
    }

    // Branch-free masked update. exp(-inf - SHIFT) == 0 kills masked lanes.
#pragma unroll
    for (int sub = 0; sub < 2; ++sub) {
      const v8f c = sub ? c1 : c0;
      const int j = jt + sub * 16 + ln;
      const int jc = cats[j];
#pragma unroll
      for (int r = 0; r < 8; ++r) {
        const float s = c[r] * INV_T;
        const int i = ibase + half * 8 + r;
        const bool same = (rcat[r] == jc);
        const float sneg = same ? NEG_INF : s;            // v_cndmask
        ns[r] += __expf(sneg - SHIFT);                    // v_exp_f32 + v_add
        const float spos = (same && (i != j)) ? s : NEG_INF;
        pm[r] = fmaxf(pm[r], spos);                       // v_max_f32
      }
    }
  }

  // Merge the 16 lane-partials of each half-wave (pure max / sum shuffles).
#pragma unroll
  for (int r = 0; r < 8; ++r) {
    float p = pm[r], sum = ns[r];
#pragma unroll
    for (int off = 1; off < 16; off <<= 1) {
      p = fmaxf(p, __shfl_xor(p, off, 32));
      sum += __shfl_xor(sum, off, 32);
    }
    if (ln == 0) {
      const int row = ibase + half * 8 + r;
      const size_t o = (size_t)blockIdx.y * BATCH + row;
      pm_o[o] = p;
      ns_o[o] = sum;
    }
  }
}

// Merge the NSPLIT column-range partials per row, compute per-row loss,
// accumulate (loss sum, valid count).
__global__ __launch_bounds__(256) void combine_rows(const float* __restrict__ pm_o,
                                                    const float* __restrict__ ns_o,
                                                    float* __restrict__ acc) {
  const int row = blockIdx.x * blockDim.x + threadIdx.x;
  if (row >= BATCH) return;
  float p = NEG_INF, sum = 0.0f;
  for (int s = 0; s < NSPLIT; ++s) {
    const size_t o = (size_t)s * BATCH + row;
    p = fmaxf(p, pm_o[o]);
    sum += ns_o[o];
  }
  const bool validr = (p > NEG_INF) && (sum > 0.0f);
  if (validr) {
    // lse = SHIFT + log(exp(p - SHIFT) + sum); loss = lse - p
    const float lse = SHIFT + logf(expf(p - SHIFT) + sum);
    atomicAdd(&acc[0], lse - p);
    atomicAdd(&acc[1], 1.0f);
  }
}

__global__ void finalize(const float* __restrict__ acc, float* __restrict__ out) {
  out[0] = (acc[1] > 0.0f) ? acc[0] / acc[1] : 0.0f;
}

extern "C" void kernel_launch(void* const* d_in, const int* in_sizes, int n_in,
                              void* d_out, int out_size, void* d_ws, size_t ws_size,
                              hipStream_t stream) {
  (void)in_sizes; (void)n_in; (void)out_size; (void)ws_size;
  const float* emb = (const float*)d_in[0];
  const int* cats  = (const int*)d_in[1];
  // d_in[2] (font_labels) is unused by the reference loss.
  float* out = (float*)d_out;

  char* ws = (char*)d_ws;
  float* en = (float*)ws;                          // 8192*256 f32 = 8 MB
  size_t off = (size_t)BATCH * DIM * sizeof(float);
  float* acc = (float*)(ws + off);                 // 2 floats (padded to 256B)
  off += 256;
  float* pm_o = (float*)(ws + off);                // NSPLIT*BATCH floats
  off += (size_t)NSPLIT * BATCH * sizeof(float);
  float* ns_o = (float*)(ws + off);

  init_acc<<<1, 1, 0, stream>>>(acc);
  normalize_rows<<<BATCH / 8, 256, 0, stream>>>(emb, en);
  dim3 grid(BATCH / ROWS_PER_BLOCK, NSPLIT);
  sim_online<<<grid, THREADS, 0, stream>>>(en, cats, pm_o, ns_o);
  combine_rows<<<BATCH / 256, 256, 0, stream>>>(pm_o, ns_o, acc);
  finalize<<<1, 1, 0, stream>>>(acc, out);
}